// BoxConv2d_67336497266993
// MI455X (gfx1250) — compile-verified
//
#include <hip/hip_runtime.h>
#include <cstdint>

// Problem constants (match reference)
constexpr int Bc = 4, Cc = 32, FNc = 4, Hh = 256, Ww = 256;
constexpr int P = 257;                 // padded integral pitch (H+1 = W+1)

typedef __attribute__((ext_vector_type(2))) float v2f;
typedef __attribute__((ext_vector_type(8))) float v8f;

#define USE_ASYNC_LDS 1

// ---------------------------------------------------------------------------
// Kernel 1: per-plane 2-D integral image, prefix scans done on the matrix core
// cumsum(X) along a 16-block  ==  X(16x16) * T(16x16, upper-tri ones), done as
// four chained V_WMMA_F32_16X16X4_F32; carry between blocks via wave shuffles.
// One workgroup (512 thr = 16 waves) per (b,c) plane; whole plane in LDS.
// ---------------------------------------------------------------------------
__global__ void integral_image_wmma(const float* __restrict__ x,
                                    float* __restrict__ I) {
  extern __shared__ float lds[];               // 256 * 257 floats (263,168 B)
  const int tid  = threadIdx.x;                // 0..511
  const int lane = tid & 31;
  const int wave = tid >> 5;                   // 0..15
  const int hi   = lane >> 4;                  // 0/1 (lane half)
  const int ln   = lane & 15;                  // 0..15

  const int plane = blockIdx.x;                // b*C + c
  const float* xp = x + (size_t)plane * (Hh * Ww);
  float* Ip       = I + (size_t)plane * (P * P);

  // --- load plane into LDS (pitch 257 kills column-scan bank conflicts) ---
  for (int k = tid; k < Hh * Ww; k += 512) {
    const int i = k >> 8, j = k & 255;
    lds[i * P + j] = xp[k];
  }
  __syncthreads();

  // --- stage 1: cumsum along W.  Wave w owns rows [16w, 16w+16) ---
  {
    const int rowbase = wave * 16;
    float carry[8];
#pragma unroll
    for (int v = 0; v < 8; ++v) carry[v] = 0.0f;

    for (int blk = 0; blk < 16; ++blk) {
      const int cb = blk * 16;
      v8f acc = {};
#pragma unroll
      for (int q = 0; q < 4; ++q) {
        const int k0 = 4 * q + 2 * hi;         // local K index for VGPR0
        v2f a, b;
        a.x = lds[(rowbase + ln) * P + cb + k0];       // A = X block (16x4)
        a.y = lds[(rowbase + ln) * P + cb + k0 + 1];
        b.x = (k0     <= ln) ? 1.0f : 0.0f;            // B = T rows (4x16)
        b.y = (k0 + 1 <= ln) ? 1.0f : 0.0f;
        acc = __builtin_amdgcn_wmma_f32_16x16x4_f32(
            false, a, false, b, (short)0, acc, false, false);
      }
#pragma unroll
      for (int v = 0; v < 8; ++v) {
        float y = acc[v] + carry[v];                   // add running prefix
        carry[v] = __shfl(y, lane | 15, 32);           // N=15 col -> new carry
        lds[(rowbase + v + 8 * hi) * P + cb + ln] = y; // (M, N) scatter back
      }
    }
  }
  __syncthreads();

  // --- stage 2: cumsum along H.  Wave w owns cols [16w, 16w+16) ---
  // Y = L(16x16 lower-tri ones) * X ; A = L generated in registers.
  {
    const int colbase = wave * 16;
    float cy = 0.0f;                                   // per-column carry
    for (int blk = 0; blk < 16; ++blk) {
      const int rb = blk * 16;
      v8f acc = {};
#pragma unroll
      for (int q = 0; q < 4; ++q) {
        const int k0 = 4 * q + 2 * hi;
        v2f a, b;
        a.x = (k0     <= ln) ? 1.0f : 0.0f;            // L[m=ln][k0]
        a.y = (k0 + 1 <= ln) ? 1.0f : 0.0f;
        b.x = lds[(rb + k0)     * P + colbase + ln];   // B = X rows (4x16)
        b.y = lds[(rb + k0 + 1) * P + colbase + ln];
        acc = __builtin_amdgcn_wmma_f32_16x16x4_f32(
            false, a, false, b, (short)0, acc, false, false);
      }
      float y[8];
#pragma unroll
      for (int v = 0; v < 8; ++v) y[v] = acc[v] + cy;
      cy = __shfl(y[7], 16 + ln, 32);                  // row M=15 -> new carry
#pragma unroll
      for (int v = 0; v < 8; ++v)
        lds[(rb + v + 8 * hi) * P + colbase + ln] = y[v];
    }
  }
  __syncthreads();

  // --- write zero-padded integral image (row 0 / col 0 are zeros) ---
  for (int k = tid; k < P * P; k += 512) {
    const int i = k / P, j = k - i * P;
    Ip[k] = (i == 0 || j == 0) ? 0.0f : lds[(i - 1) * P + (j - 1)];
  }
}

// ---------------------------------------------------------------------------
// Kernel 2: fractional box sampling. One block per (b, c*FN+f, h); 256 threads
// = one output row. The 4 integral rows needed by the whole row (~4 KB) are
// staged into LDS with gfx1250 async global->LDS loads; all 16 gathers/pixel
// then hit LDS. Memory-bound: dominated by the 134 MB of output stores.
// ---------------------------------------------------------------------------
__device__ __forceinline__ void splitc(float coord, int dim, int& i, float& fr) {
  float c  = fminf(fmaxf(coord, 0.0f), (float)dim);
  float fi = fminf(floorf(c), (float)(dim - 1));
  i  = (int)fi;
  fr = c - fi;
}

__global__ void boxconv_sample(const float* __restrict__ I,
                               const float* __restrict__ x_min,
                               const float* __restrict__ x_max,
                               const float* __restrict__ y_min,
                               const float* __restrict__ y_max,
                               float* __restrict__ out) {
  __shared__ float rowbuf[4 * P];
  const int h   = blockIdx.x;
  const int cf  = blockIdx.y;            // c*FN + f
  const int b   = blockIdx.z;
  const int c   = cf >> 2;               // FN == 4
  const int tid = threadIdx.x;           // w

  const float* Ip = I + (size_t)(b * Cc + c) * (P * P);

  // Row-uniform vertical coordinates (scalar path)
  int ixl, ixh; float fxl, fxh;
  splitc((float)h + x_min[cf],        Hh, ixl, fxl);
  splitc((float)h + x_max[cf] + 1.0f, Hh, ixh, fxh);

  const int rows[4] = { ixl, ixl + 1, ixh, ixh + 1 };
#pragma unroll
  for (int r = 0; r < 4; ++r) {
    for (int j = tid; j < P; j += 256) {
#if USE_ASYNC_LDS
      asm volatile("global_load_async_to_lds_b32 %0, %1, off"
                   :
                   : "v"((unsigned)(uintptr_t)&rowbuf[r * P + j]),
                     "v"(Ip + rows[r] * P + j)
                   : "memory");
#else
      rowbuf[r * P + j] = Ip[rows[r] * P + j];
#endif
    }
  }
#if USE_ASYNC_LDS
  asm volatile("s_wait_asynccnt 0x0" ::: "memory");
#endif
  __syncthreads();

  // Per-pixel horizontal coordinates
  int iyl, iyh; float fyl, fyh;
  splitc((float)tid + y_min[cf],        Ww, iyl, fyl);
  splitc((float)tid + y_max[cf] + 1.0f, Ww, iyh, fyh);

  auto samp = [&](int rb, float fx, int iy, float fy) -> float {
    const float v00 = rowbuf[rb * P + iy];
    const float v01 = rowbuf[rb * P + iy + 1];
    const float v10 = rowbuf[(rb + 1) * P + iy];
    const float v11 = rowbuf[(rb + 1) * P + iy + 1];
    const float a0 = v00 + fx * (v10 - v00);
    const float a1 = v01 + fx * (v11 - v01);
    return a0 + fy * (a1 - a0);
  };

  const float val = samp(2, fxh, iyh, fyh) - samp(0, fxl, iyh, fyh)
                  - samp(2, fxh, iyl, fyl) + samp(0, fxl, iyl, fyl);

  out[(((size_t)b * (Cc * FNc) + cf) * Hh + h) * Ww + tid] = val;
}

// ---------------------------------------------------------------------------
extern "C" void kernel_launch(void* const* d_in, const int* in_sizes, int n_in,
                              void* d_out, int out_size, void* d_ws, size_t ws_size,
                              hipStream_t stream) {
  (void)in_sizes; (void)n_in; (void)out_size; (void)ws_size;
  const float* x    = (const float*)d_in[0];
  const float* xmin = (const float*)d_in[1];
  const float* xmax = (const float*)d_in[2];
  const float* ymin = (const float*)d_in[3];
  const float* ymax = (const float*)d_in[4];

  float* I = (float*)d_ws;   // B*C*257*257 f32 = 33.8 MB of workspace

  // Stage 1: integral images (one workgroup per plane, whole plane in LDS)
  hipLaunchKernelGGL(integral_image_wmma, dim3(Bc * Cc), dim3(512),
                     (size_t)Hh * P * sizeof(float), stream, x, I);

  // Stage 2: box sampling (one block per output row)
  dim3 grid(Hh, Cc * FNc, Bc);
  hipLaunchKernelGGL(boxconv_sample, grid, dim3(Ww), 0, stream,
                     I, xmin, xmax, ymin, ymax, (float*)d_out);
}